// BackpropagationBasedFiltering_3590592659458
// MI455X (gfx1250) — compile-verified
//
#include <hip/hip_runtime.h>
#include <math.h>

#define T_STEPS  200000
#define NLIMB    49
#define NPTS     50
#define NBONE    20
#define KPAD     52           // 13 * 4
#define NPAD     64           // 4 * 16
#define KTILES   13
#define FEPS     1e-10f

// workspace layout (floats)
#define WS_B     0            // Bt[n*KPAD + k] = PATH_T[k][n], 64*52
#define WS_L     (NPAD*KPAD)  // Lpad[52]
#define WS_REG1  (WS_L + KPAD)
#define WS_PART  (WS_REG1 + 12)
#define NB_LOSS  2048

typedef float v2f __attribute__((ext_vector_type(2)));
typedef float v8f __attribute__((ext_vector_type(8)));
typedef int   g4i __attribute__((vector_size(16)));   // int4 as GCC vector

#if defined(__has_builtin)
#if __has_builtin(__builtin_amdgcn_global_load_async_to_lds_b128) && \
    __has_builtin(__builtin_amdgcn_s_wait_asynccnt)
#define HAVE_ASYNC_LDS 1
#endif
#endif

__device__ __forceinline__ float path_val(int n, int k) {
  // 1.0 if limb k (edge into node k+1) lies on the root->node-n path
  if (n >= NPTS || k >= NLIMB) return 0.0f;
  int target = k + 1;
  int c = n;
  while (c > 0) {
    if (c == target) return 1.0f;
    c = (c - 1) >> 1;
  }
  return 0.0f;
}

__global__ void __launch_bounds__(256)
precompute_kernel(const float* __restrict__ lines, float* __restrict__ ws) {
  int idx = blockIdx.x * 256 + threadIdx.x;
  if (idx < NPAD * KPAD) {
    int n = idx / KPAD, k = idx % KPAD;
    ws[WS_B + idx] = path_val(n, k);
  } else if (idx < NPAD * KPAD + KPAD) {
    int k = idx - NPAD * KPAD;
    ws[WS_L + k] = (k < NLIMB) ? expf(lines[k % NBONE]) : 0.0f;
  } else if (idx == NPAD * KPAD + KPAD) {
    float s = 0.0f;
    for (int i = 0; i < NBONE; ++i) s += expf(lines[i]);
    ws[WS_REG1] = s;
  }
}

__device__ __forceinline__ v8f wmma4f32(v2f a, v2f b, v8f c) {
  return __builtin_amdgcn_wmma_f32_16x16x4_f32(
      /*neg_a=*/false, a, /*neg_b=*/false, b,
      /*c_mod=*/(short)0, c, /*reuse_a=*/false, /*reuse_b=*/false);
}

__device__ __forceinline__ void kstep(const float* sB, int ncol, int kk,
                                      v2f Ax, v2f Ay, v2f Az,
                                      v8f accx[4], v8f accy[4], v8f accz[4]) {
#pragma unroll
  for (int nt = 0; nt < 4; ++nt) {
    const int nb = (nt * 16 + ncol) * KPAD + kk;
    v2f B; B[0] = sB[nb]; B[1] = sB[nb + 1];
    accx[nt] = wmma4f32(Ax, B, accx[nt]);
    accy[nt] = wmma4f32(Ay, B, accy[nt]);
    accz[nt] = wmma4f32(Az, B, accz[nt]);
  }
}

__global__ void __launch_bounds__(128)
fk_wmma_kernel(const float* __restrict__ ax, const float* __restrict__ ay,
               const float* __restrict__ az,
               const float* __restrict__ rx, const float* __restrict__ ry,
               const float* __restrict__ rz,
               const float* __restrict__ ws,
               float* __restrict__ outx, float* __restrict__ outy,
               float* __restrict__ outz) {
  __shared__ float4 sB4[(NPAD * KPAD) / 4];   // 832 float4
  __shared__ float4 sL4[KPAD / 4];            // 13 float4
  const float* sB = (const float*)sB4;
  const float* sL = (const float*)sL4;
  const int tid = threadIdx.x;

#ifdef HAVE_ASYNC_LDS
  {
    // Async global->LDS staging of the path matrix + bone lengths
    // (ASYNCcnt-tracked). Builtin takes (AS1 int4* src, AS3 int4* dst, imm, imm).
    typedef __attribute__((address_space(1))) g4i* gv4p;
    typedef __attribute__((address_space(3))) g4i* lv4p;
    gv4p gB = (gv4p)(ws + WS_B);
    gv4p gL = (gv4p)(ws + WS_L);
    lv4p lB = (lv4p)sB4;
    lv4p lL = (lv4p)sL4;
    for (int i = tid; i < (NPAD * KPAD) / 4; i += 128)
      __builtin_amdgcn_global_load_async_to_lds_b128(gB + i, lB + i, 0, 0);
    if (tid < KPAD / 4)
      __builtin_amdgcn_global_load_async_to_lds_b128(gL + tid, lL + tid, 0, 0);
    __builtin_amdgcn_s_wait_asynccnt(0);
  }
#else
  {
    const float4* gB = (const float4*)(ws + WS_B);
    const float4* gL = (const float4*)(ws + WS_L);
    for (int i = tid; i < (NPAD * KPAD) / 4; i += 128) sB4[i] = gB[i];
    if (tid < KPAD / 4) sL4[tid] = gL[tid];
  }
#endif
  __syncthreads();

  const int wave  = tid >> 5;
  const int lane  = tid & 31;
  const int mtile = blockIdx.x * 4 + wave;           // 12500 M-tiles exactly
  const int mrow  = mtile * 16 + (lane & 15);        // A-fragment row
  const int khalf = (lane >> 4) * 2;                 // 0 or 2
  const int ncol  = lane & 15;

  v8f accx[4], accy[4], accz[4];
#pragma unroll
  for (int nt = 0; nt < 4; ++nt) {
#pragma unroll
    for (int i = 0; i < 8; ++i) { accx[nt][i] = 0.0f; accy[nt][i] = 0.0f; accz[nt][i] = 0.0f; }
  }

  // k-tiles 0..11: kk+1 <= 47 < 49, no bounds checks needed.
  for (int kt = 0; kt < KTILES - 1; ++kt) {
    const int kk = kt * 4 + khalf;
    const size_t o = (size_t)mrow * NLIMB + kk;
    v2f Ax, Ay, Az;
#pragma unroll
    for (int j = 0; j < 2; ++j) {
      const float a0 = ax[o + j], a1 = ay[o + j], a2 = az[o + j];
      const float nrm = sqrtf(a0 * a0 + a1 * a1 + a2 * a2) + FEPS;
      const float s = sL[kk + j] / nrm;
      Ax[j] = a0 * s; Ay[j] = a1 * s; Az[j] = a2 * s;
    }
    kstep(sB, ncol, kk, Ax, Ay, Az, accx, accy, accz);
  }

  // Peeled last k-tile (kt=12): only k=48 is in range (lanes with khalf==0, j==0).
  {
    const int kk = (KTILES - 1) * 4 + khalf;         // 48 or 50
    v2f Ax, Ay, Az;
#pragma unroll
    for (int j = 0; j < 2; ++j) {
      const int k = kk + j;
      float vx = 0.0f, vy = 0.0f, vz = 0.0f;
      if (k < NLIMB) {
        const size_t o = (size_t)mrow * NLIMB + k;
        const float a0 = ax[o], a1 = ay[o], a2 = az[o];
        const float nrm = sqrtf(a0 * a0 + a1 * a1 + a2 * a2) + FEPS;
        const float s = sL[k] / nrm;
        vx = a0 * s; vy = a1 * s; vz = a2 * s;
      }
      Ax[j] = vx; Ay[j] = vy; Az[j] = vz;
    }
    kstep(sB, ncol, kk, Ax, Ay, Az, accx, accy, accz);
  }

  // C/D layout: element (vgpr v, lane l) holds M = v + 8*(l>=16), N = l&15
  const int mbase = mtile * 16 + 8 * (lane >> 4);
#pragma unroll
  for (int v = 0; v < 8; ++v) {
    const int row = mbase + v;
    const float xr = rx[row], yr = ry[row], zr = rz[row];
#pragma unroll
    for (int nt = 0; nt < 4; ++nt) {
      const int n = nt * 16 + ncol;
      if (n < NPTS) {
        const size_t o = (size_t)row * NPTS + n;
        outx[o] = accx[nt][v] + xr;
        outy[o] = accy[nt][v] + yr;
        outz[o] = accz[nt][v] + zr;
      }
    }
  }
}

__global__ void __launch_bounds__(256)
loss_kernel(const float* __restrict__ x, const float* __restrict__ y,
            const float* __restrict__ z,
            const float* __restrict__ tarx, const float* __restrict__ tary,
            const float* __restrict__ w, float* __restrict__ ws) {
  const long total = (long)T_STEPS * NPTS;
  float lacc = 0.0f, racc = 0.0f;
  for (long idx = (long)blockIdx.x * 256 + threadIdx.x; idx < total;
       idx += (long)gridDim.x * 256) {
    const float xv = x[idx], yv = y[idx];
    const float dx = xv - tarx[idx];
    const float dy = yv - tary[idx];
    lacc += w[idx] * (dx * dx + dy * dy);
    if (idx < total - NPTS) {
      const float zv = z[idx];
      const float tx = xv - x[idx + NPTS];
      const float ty = yv - y[idx + NPTS];
      const float tz = zv - z[idx + NPTS];
      racc += tx * tx + ty * ty + tz * tz;
    }
  }
  __shared__ float sl[256], sr[256];
  sl[threadIdx.x] = lacc; sr[threadIdx.x] = racc;
  __syncthreads();
  for (int s = 128; s > 0; s >>= 1) {
    if ((int)threadIdx.x < s) {
      sl[threadIdx.x] += sl[threadIdx.x + s];
      sr[threadIdx.x] += sr[threadIdx.x + s];
    }
    __syncthreads();
  }
  if (threadIdx.x == 0) {
    ws[WS_PART + blockIdx.x] = sl[0];
    ws[WS_PART + NB_LOSS + blockIdx.x] = sr[0];
  }
}

__global__ void __launch_bounds__(256)
final_kernel(const float* __restrict__ ws, float* __restrict__ out_loss) {
  __shared__ double sl[256], sr[256];
  double lacc = 0.0, racc = 0.0;
  for (int i = threadIdx.x; i < NB_LOSS; i += 256) {
    lacc += (double)ws[WS_PART + i];
    racc += (double)ws[WS_PART + NB_LOSS + i];
  }
  sl[threadIdx.x] = lacc; sr[threadIdx.x] = racc;
  __syncthreads();
  for (int s = 128; s > 0; s >>= 1) {
    if ((int)threadIdx.x < s) {
      sl[threadIdx.x] += sl[threadIdx.x + s];
      sr[threadIdx.x] += sr[threadIdx.x + s];
    }
    __syncthreads();
  }
  if (threadIdx.x == 0) {
    const double loss = sl[0] / ((double)T_STEPS * NPTS);
    const double reg2 = sr[0] / ((double)(T_STEPS - 1) * NPTS);
    const double reg1 = (double)ws[WS_REG1];
    out_loss[0] = (float)(loss + 0.001 * reg1 + 0.1 * reg2);
  }
}

extern "C" void kernel_launch(void* const* d_in, const int* in_sizes, int n_in,
                              void* d_out, int out_size, void* d_ws, size_t ws_size,
                              hipStream_t stream) {
  const float* lines = (const float*)d_in[0];
  const float* rx    = (const float*)d_in[1];
  const float* ry    = (const float*)d_in[2];
  const float* rz    = (const float*)d_in[3];
  const float* ax    = (const float*)d_in[4];
  const float* ay    = (const float*)d_in[5];
  const float* az    = (const float*)d_in[6];
  const float* tarx  = (const float*)d_in[7];
  const float* tary  = (const float*)d_in[8];
  const float* w     = (const float*)d_in[9];

  float* ws   = (float*)d_ws;
  float* outx = (float*)d_out;
  float* outy = outx + (size_t)T_STEPS * NPTS;
  float* outz = outy + (size_t)T_STEPS * NPTS;
  float* outl = outz + (size_t)T_STEPS * NPTS;

  precompute_kernel<<<(NPAD * KPAD + KPAD + 1 + 255) / 256, 256, 0, stream>>>(lines, ws);
  fk_wmma_kernel<<<(T_STEPS / 16) / 4, 128, 0, stream>>>(ax, ay, az, rx, ry, rz,
                                                         ws, outx, outy, outz);
  loss_kernel<<<NB_LOSS, 256, 0, stream>>>(outx, outy, outz, tarx, tary, w, ws);
  final_kernel<<<1, 256, 0, stream>>>(ws, outl);
}